// QuantLinearLUT_25769804260
// MI455X (gfx1250) — compile-verified
//
#include <hip/hip_runtime.h>

// ---------------------------------------------------------------------------
// QuantLinearLUT on MI455X (gfx1250, wave32)
//
// Roofline: qidx is 256MB of int32 -> bandwidth floor ~11.3us @ 23.3TB/s.
// Dense math (8x8192x8192 = 1.07 GFLOP fp32) goes to the matrix pipe via
// V_WMMA_F32_16X16X4_F32 so the VALU is free for the LUT dequant gather,
// keeping the kernel on the memory roof instead of VALU-issue-bound.
// Sparse CSR part (1M nnz) is noise; handled by a vectorized per-column kernel.
// Determinism: K-split partial sums live in d_ws, summed by a reduce kernel
// (no float atomics).
// ---------------------------------------------------------------------------

typedef float v2f __attribute__((ext_vector_type(2)));
typedef float v8f __attribute__((ext_vector_type(8)));

#define B_      8
#define IF_     8192
#define OF_     8192
#define LUTN    8
#define NSPLIT  8                    // K-dimension split for occupancy
#define KCHUNK  (IF_ / NSPLIT)       // 1024 rows of qidx per workgroup
#define LUT_STRIDE 9                 // pad 8->9 words: lane banks distinct mod 64

// ---------------------------------------------------------------------------
// Dense LUT-GEMM: y_partial[split][b][o] = sum_{k in split} x[b][k]*LUT[o][qidx[k][o]]
// Wave owns 16 output columns (one WMMA N-tile); workgroup = 8 waves = 128 cols.
// WMMA f32 16x16x4 operand layout (wave32):
//   A (16x4, M=batch): lane L<16 -> a={x[L][k0],x[L][k0+1]}, lanes 16-31 -> k0+2/k0+3
//   B (4x16, N=cols):  lane L<16 -> b={W[k0][nL],W[k0+1][nL]}, lanes 16-31 -> rows k0+2/3
//   C/D (16x16):       VGPR v, lanes 0-15 -> (M=v, N=lane)   [M=v is batch row 0..7]
// ---------------------------------------------------------------------------
__global__ __launch_bounds__(256) void dense_lut_wmma_kernel(
    const float* __restrict__ x,      // (B, IF) row-major
    const int*   __restrict__ qidx,   // (IF, OF) row-major
    const float* __restrict__ lut,    // (OF, LUTN) row-major
    float*       __restrict__ partial)// (NSPLIT, B, OF)
{
    __shared__ float slut[128 * LUT_STRIDE];

    const int nbase  = blockIdx.x * 128;       // first column of this workgroup
    const int ksplit = blockIdx.y;
    const int kstart = ksplit * KCHUNK;
    const int tid    = threadIdx.x;

    // Cooperative LUT stage: 128 cols x 8 entries, contiguous in global memory.
    for (int e = tid; e < 128 * LUTN; e += 256) {
        const int c = e >> 3, q = e & 7;
        slut[c * LUT_STRIDE + q] = lut[(size_t)(nbase)*LUTN + e];
    }
    __syncthreads();

    const int wave = tid >> 5;
    const int lane = tid & 31;
    const int half = lane >> 4;          // 0: K rows k0,k0+1   1: rows k0+2,k0+3
    const int nl   = lane & 15;
    const int cwg  = wave * 16 + nl;     // column index within workgroup
    const int n    = nbase + cwg;        // global output column for B operand
    const int row_off = half * 2;
    const float* lutp = &slut[cwg * LUT_STRIDE];

    // A operand: batch row = nl (rows 8..15 are zero padding of the M=16 tile)
    const int brow = nl;
    const int koff = half * 2;
    const float* xrow = x + (size_t)(brow < B_ ? brow : 0) * IF_;

    v8f acc = {};
    const int steps = KCHUNK / 4;        // 256 WMMAs per wave

#pragma unroll 2
    for (int s = 0; s < steps; ++s) {
        const int k0 = kstart + s * 4;

        // --- B operand: dequantize W[k][n] via LDS LUT ---
        const int q0 = qidx[(k0 + row_off) * OF_ + n];
        const int q1 = qidx[(k0 + row_off + 1) * OF_ + n];
        v2f bv;
        bv.x = lutp[q0];
        bv.y = lutp[q1];

        // --- A operand: x tile (L2-resident, 256KB total); mask rows >= B ---
        v2f av = *(const v2f*)(xrow + k0 + koff);
        av.x = (brow < B_) ? av.x : 0.0f;
        av.y = (brow < B_) ? av.y : 0.0f;

        // --- prefetch qidx 16 steps (64 rows) ahead, clamped in-range ---
        int kp = k0 + 64;
        kp = (kp < kstart + KCHUNK - 2) ? kp : k0;
        __builtin_prefetch(&qidx[(kp + row_off) * OF_ + n], 0, 1);

        // D = A x B + C   (matrix pipe; 1024 MACs, 512 useful M rows)
        acc = __builtin_amdgcn_wmma_f32_16x16x4_f32(
            /*neg_a=*/false, av, /*neg_b=*/false, bv,
            /*c_mod=*/(short)0, acc, /*reuse_a=*/false, /*reuse_b=*/false);
    }

    // Store: VGPR v, lanes 0-15 hold (M=v, N=nl); M=v covers batch rows 0..7.
    if (half == 0) {
        float* p = partial + (size_t)ksplit * (B_ * OF_);
#pragma unroll
        for (int v = 0; v < 8; ++v)
            p[v * OF_ + n] = acc[v];
    }
}

// ---------------------------------------------------------------------------
// Transpose x (B,IF) -> xT (IF,B): makes the sparse gather x[:,col] one
// contiguous 32B fetch instead of 8 loads strided by 32KB.
// ---------------------------------------------------------------------------
__global__ __launch_bounds__(256) void xT_kernel(
    const float* __restrict__ x, float* __restrict__ xT)
{
    const int i = blockIdx.x * 256 + threadIdx.x;   // 0..IF-1
#pragma unroll
    for (int b = 0; b < B_; ++b)
        xT[(size_t)i * B_ + b] = x[(size_t)b * IF_ + i];
}

// ---------------------------------------------------------------------------
// Sparse CSR contribution: one thread per output column o.
// y_sp[b][o] = sum_j vals[j] * x[b][cols[j]],  j in rows[o]..rows[o+1]
// ---------------------------------------------------------------------------
__global__ __launch_bounds__(256) void sparse_kernel(
    const float* __restrict__ xT,     // (IF, B)
    const int*   __restrict__ rows,
    const int*   __restrict__ cols,
    const float* __restrict__ vals,
    float*       __restrict__ outp)   // (B, OF)
{
    const int o  = blockIdx.x * 256 + threadIdx.x;  // 0..OF-1
    const int r0 = rows[o];
    const int r1 = rows[o + 1];

    float a0 = 0.f, a1 = 0.f, a2 = 0.f, a3 = 0.f;
    float a4 = 0.f, a5 = 0.f, a6 = 0.f, a7 = 0.f;

    int j = r0;
    for (; j + 3 < r1; j += 4) {                     // rows[] are 128-aligned
        const int4   c4 = *(const int4*)(cols + j);
        const float4 v4 = *(const float4*)(vals + j);
        const int cc[4] = {c4.x, c4.y, c4.z, c4.w};
        const float vv[4] = {v4.x, v4.y, v4.z, v4.w};
#pragma unroll
        for (int t = 0; t < 4; ++t) {
            const float4* p = (const float4*)(xT + (size_t)cc[t] * B_);
            const float4 p0 = p[0];
            const float4 p1 = p[1];
            const float v = vv[t];
            a0 += v * p0.x; a1 += v * p0.y; a2 += v * p0.z; a3 += v * p0.w;
            a4 += v * p1.x; a5 += v * p1.y; a6 += v * p1.z; a7 += v * p1.w;
        }
    }
    for (; j < r1; ++j) {                            // generic tail
        const int   c = cols[j];
        const float v = vals[j];
        const float* p = xT + (size_t)c * B_;
        a0 += v * p[0]; a1 += v * p[1]; a2 += v * p[2]; a3 += v * p[3];
        a4 += v * p[4]; a5 += v * p[5]; a6 += v * p[6]; a7 += v * p[7];
    }

    outp[0 * OF_ + o] = a0; outp[1 * OF_ + o] = a1;
    outp[2 * OF_ + o] = a2; outp[3 * OF_ + o] = a3;
    outp[4 * OF_ + o] = a4; outp[5 * OF_ + o] = a5;
    outp[6 * OF_ + o] = a6; outp[7 * OF_ + o] = a7;
}

// ---------------------------------------------------------------------------
// Final reduce: out = sum(dense partials over NSPLIT) + sparse partial.
// ---------------------------------------------------------------------------
__global__ __launch_bounds__(256) void reduce_kernel(
    const float* __restrict__ wsf, float* __restrict__ out)
{
    const int idx = blockIdx.x * 256 + threadIdx.x;  // 0..B*OF-1
    float s = wsf[(size_t)NSPLIT * B_ * OF_ + idx];  // sparse slice
#pragma unroll
    for (int p = 0; p < NSPLIT; ++p)
        s += wsf[(size_t)p * B_ * OF_ + idx];
    out[idx] = s;
}

// ---------------------------------------------------------------------------
// Launch. Input order: x, qidx, lookup_table, rows, cols, vals.
// Workspace layout (floats): [NSPLIT*B*OF dense partials][B*OF sparse][IF*B xT]
// = 2.5 MB total.
// ---------------------------------------------------------------------------
extern "C" void kernel_launch(void* const* d_in, const int* in_sizes, int n_in,
                              void* d_out, int out_size, void* d_ws, size_t ws_size,
                              hipStream_t stream) {
    const float* x    = (const float*)d_in[0];
    const int*   qidx = (const int*)  d_in[1];
    const float* lut  = (const float*)d_in[2];
    const int*   rows = (const int*)  d_in[3];
    const int*   cols = (const int*)  d_in[4];
    const float* vals = (const float*)d_in[5];
    float* out = (float*)d_out;

    float* wsf      = (float*)d_ws;
    float* dense_p  = wsf;                                  // NSPLIT * B * OF
    float* sparse_p = wsf + (size_t)NSPLIT * B_ * OF_;      // B * OF
    float* xT       = sparse_p + (size_t)B_ * OF_;          // IF * B

    xT_kernel<<<IF_ / 256, 256, 0, stream>>>(x, xT);
    dense_lut_wmma_kernel<<<dim3(OF_ / 128, NSPLIT), 256, 0, stream>>>(
        x, qidx, lut, dense_p);
    sparse_kernel<<<OF_ / 256, 256, 0, stream>>>(xT, rows, cols, vals, sparse_p);
    reduce_kernel<<<(B_ * OF_) / 256, 256, 0, stream>>>(wsf, out);
}